// IoULoss_63170378990199
// MI455X (gfx1250) — compile-verified
//
#include <hip/hip_runtime.h>
#include <stdint.h>

// ---------------------------------------------------------------------------
// IoU loss: out[j] = 1 - max_i IoU(fix(pred_i), fix(true_j)),  N = M = 8192.
// Compute-bound (~1.2 G VALU ops, ~256 KB total traffic -> L2 resident).
// No sum-of-products structure -> WMMA inapplicable (IoU is min/max/clamp +
// a MAX-reduction). CDNA5-specific paths used instead: async global->LDS
// copy (ASYNCcnt), global_prefetch_b8, wave32 tiling, LDS broadcast loop.
// ---------------------------------------------------------------------------

#define TILE 256   // boxes per tile == threads per block (8 wave32s)

#if defined(__has_builtin)
#  if __has_builtin(__builtin_amdgcn_global_load_async_to_lds_b128)
#    define USE_ASYNC_LDS 1
#  endif
#endif

// Builtin signature (from hipcc diagnostic): pointer-to-v4i with address
// spaces: src in addrspace(1) (global), dst in addrspace(3) (LDS).
typedef int v4i __attribute__((vector_size(16)));
typedef __attribute__((address_space(1))) v4i* gptr_v4i;
typedef __attribute__((address_space(3))) v4i* lptr_v4i;

__device__ __forceinline__ float4 fix_box(float4 b) {
    // matches _fix_box_coordinates with EPS = 1.0
    float x1 = fminf(b.x, b.z);
    float y1 = fminf(b.y, b.w);
    float x2 = fmaxf(b.x, b.z);
    float y2 = fmaxf(b.y, b.w);
    x2 = fmaxf(x2, x1 + 1.0f);
    y2 = fmaxf(y2, y1 + 1.0f);
    return make_float4(x1, y1, x2, y2);
}

__global__ __launch_bounds__(TILE)
void iou_tile_max_kernel(const float4* __restrict__ pred,
                         const float4* __restrict__ tru,
                         unsigned int* __restrict__ maxbuf,
                         int N, int M) {
    __shared__ float4 sbox[TILE];

    const int t     = threadIdx.x;
    const int j     = blockIdx.x * TILE + t;       // true-box (output) index
    const int ibase = blockIdx.y * TILE;           // pred-tile base
    const int tileN = min(TILE, N - ibase);

    // gfx1250 prefetch of this thread's true box (global_prefetch_b8).
    if (j < M) __builtin_prefetch(tru + j, 0, 1);

    // ---- Stage pred tile into LDS --------------------------------------
#if USE_ASYNC_LDS
    if (t < tileN) {
        __builtin_amdgcn_global_load_async_to_lds_b128(
            (gptr_v4i)(pred + ibase + t),
            (lptr_v4i)&sbox[t],
            /*imm offset=*/0, /*cpol=*/0);
    }
#  if __has_builtin(__builtin_amdgcn_s_wait_asynccnt)
    __builtin_amdgcn_s_wait_asynccnt(0);
#  else
    asm volatile("s_wait_asynccnt 0x0" ::: "memory");
#  endif
    __syncthreads();
    if (t < tileN) sbox[t] = fix_box(sbox[t]);     // fix coords in place
    __syncthreads();
#else
    if (t < tileN) sbox[t] = fix_box(pred[ibase + t]);
    __syncthreads();
#endif

    // ---- This thread's true box (registers) ----------------------------
    float4 tb = make_float4(0.f, 0.f, 1.f, 1.f);
    if (j < M) tb = fix_box(tru[j]);
    const float tarea = (tb.z - tb.x) * (tb.w - tb.y);

    // ---- Scan pred tile; division-free running max ---------------------
    // iou_k > iou_best  <=>  inter_k * uni_best > inter_best * uni_k
    // (unions strictly positive because the EPS fix forces area >= 1).
    float bestI = 0.0f;   // intersection of current best
    float bestU = 1.0f;   // union of current best  (0/1 -> iou 0)
#pragma unroll 4
    for (int k = 0; k < tileN; ++k) {
        float4 p = sbox[k];                        // ds_load_b128 broadcast
        float iw = fmaxf(fminf(p.z, tb.z) - fmaxf(p.x, tb.x), 0.0f);
        float ih = fmaxf(fminf(p.w, tb.w) - fmaxf(p.y, tb.y), 0.0f);
        float inter = iw * ih;
        float parea = (p.z - p.x) * (p.w - p.y);
        float uni   = parea + tarea - inter;
        bool better = inter * bestU > bestI * uni;
        bestI = better ? inter : bestI;
        bestU = better ? uni   : bestU;
    }

    if (j < M) {
        float iou = bestI / bestU;                 // one divide per tile
        // IoU >= 0, so uint bit-pattern compare == float compare.
        atomicMax(maxbuf + j, __float_as_uint(iou));
    }
}

__global__ void iou_finalize_kernel(const unsigned int* __restrict__ maxbuf,
                                    float* __restrict__ out, int M) {
    int j = blockIdx.x * blockDim.x + threadIdx.x;
    if (j < M) out[j] = 1.0f - __uint_as_float(maxbuf[j]);
}

extern "C" void kernel_launch(void* const* d_in, const int* in_sizes, int n_in,
                              void* d_out, int out_size, void* d_ws, size_t ws_size,
                              hipStream_t stream) {
    const float4* pred = (const float4*)d_in[0];
    const float4* tru  = (const float4*)d_in[1];
    const int N = in_sizes[0] / 4;
    const int M = in_sizes[1] / 4;

    unsigned int* maxbuf = (unsigned int*)d_ws;    // M uints of scratch
    (void)hipMemsetAsync(maxbuf, 0, (size_t)M * sizeof(unsigned int), stream);

    dim3 grid((M + TILE - 1) / TILE, (N + TILE - 1) / TILE);
    iou_tile_max_kernel<<<grid, TILE, 0, stream>>>(pred, tru, maxbuf, N, M);

    int fb = 256;
    iou_finalize_kernel<<<(M + fb - 1) / fb, fb, 0, stream>>>(
        maxbuf, (float*)d_out, M);
}